// HybridQLSTM_65481071407792
// MI455X (gfx1250) — compile-verified
//
#include <hip/hip_runtime.h>

// ---------------------------------------------------------------------------
// HybridQLSTM, algebraically collapsed:
//   * hidden/cell states are rank-1 (broadcast gates + zero init)
//   * depth-2 RX/CNOT circuit: <Z0> == cos^2(angle[0])  (Heisenberg picture)
// so the only heavy op is inputs[65536,256] @ Wall^T[256,32]  (WMMA f32),
// then a trivial serial scalar scan and a 67MB broadcast write.
// ---------------------------------------------------------------------------

typedef float v2f __attribute__((ext_vector_type(2)));
typedef float v8f __attribute__((ext_vector_type(8)));

#define T_DIM 256
#define B_DIM 256
#define I_DIM 256
#define H_DIM 256
#define K_DIM 512          // I + H
#define NQ    8
#define NG    4
#define NROWS 32           // NG * NQ folded rows
#define MTOT  (T_DIM * B_DIM)  // 65536 GEMM rows

// workspace layout, in floats (total ~2.17M floats ~= 8.7 MB)
#define WS_WALL 0                          // [32][256]
#define WS_SQ   (NROWS * I_DIM)            // [32]
#define WS_CQ   (WS_SQ + NROWS)            // [32]
#define WS_ANGX (WS_CQ + NROWS)            // [65536][32]
#define WS_HS   (WS_ANGX + (size_t)MTOT * NROWS)  // [T][B] hs history
#define WS_CS   (WS_HS + MTOT)             // [B] final cell scalar

// ---------------------------------------------------------------------------
// 1) Fold weights: Wall[g*8+q][k] = sum_h Wq_g[q,h] * W_g[h,k]   (k < 256)
//    Sq[g*8+q]     = sum_h Wq_g[q,h] * sum_{k=256..511} W_g[h,k]
//    Cq[g*8+q]     = sum_h Wq_g[q,h] * b_g[h] + bq_g[q]
// ---------------------------------------------------------------------------
__global__ __launch_bounds__(256) void prep_kernel(
    const float* Wf, const float* bf, const float* Wfq, const float* bfq,
    const float* Wi_, const float* bi_, const float* Wiq, const float* biq,
    const float* Wu_, const float* bu_, const float* Wuq, const float* buq,
    const float* Wo_, const float* bo_, const float* Woq, const float* boq,
    float* __restrict__ wall, float* __restrict__ sq, float* __restrict__ cq)
{
    __shared__ float WqS[NQ * H_DIM];
    __shared__ float Rh[H_DIM];

    const int g = blockIdx.x;
    const int tid = threadIdx.x;
    const float* Wtab[NG]  = {Wf,  Wi_, Wu_, Wo_};
    const float* btab[NG]  = {bf,  bi_, bu_, bo_};
    const float* Wqtab[NG] = {Wfq, Wiq, Wuq, Woq};
    const float* bqtab[NG] = {bfq, biq, buq, boq};
    const float* W  = Wtab[g];
    const float* bl = btab[g];
    const float* Wq = Wqtab[g];
    const float* bq = bqtab[g];

    for (int i = tid; i < NQ * H_DIM; i += 256) WqS[i] = Wq[i];

    // row-sum of the hidden-weight half (columns I..I+H)
    float r = 0.f;
    for (int k = 0; k < H_DIM; ++k) r += W[(size_t)tid * K_DIM + I_DIM + k];
    Rh[tid] = r;
    __syncthreads();

    // Wall rows for this gate: thread tid owns column k = tid
    for (int q = 0; q < NQ; ++q) {
        float acc = 0.f;
        for (int h = 0; h < H_DIM; ++h)
            acc += WqS[q * H_DIM + h] * W[(size_t)h * K_DIM + tid];
        wall[(size_t)(g * NQ + q) * I_DIM + tid] = acc;
    }

    if (tid < NQ) {
        float s = 0.f, c = 0.f;
        for (int h = 0; h < H_DIM; ++h) {
            s += WqS[tid * H_DIM + h] * Rh[h];
            c += WqS[tid * H_DIM + h] * bl[h];
        }
        sq[g * NQ + tid] = s;
        cq[g * NQ + tid] = c + bq[tid];
    }
}

// ---------------------------------------------------------------------------
// 2) AngX[m][n] = sum_k inputs[m][k] * Wall[n][k],  m<65536, n<32, k<256
//    V_WMMA_F32_16X16X4_F32 per-wave 16x16 tile, full f32 precision.
//    A frag (16x4 f32): lanes 0-15 M=0..15; vgpr0=K(2*half), vgpr1=K(2*half+1)
//    B frag (4x16 f32): mirrored; B[k][n] = Wall[n][k] -> contiguous K pairs.
// ---------------------------------------------------------------------------
__global__ __launch_bounds__(256) void angx_gemm_kernel(
    const float* __restrict__ inp,      // [65536][256]
    const float* __restrict__ wall,     // [32][256]
    float* __restrict__ angx)           // [65536][32]
{
    const int wave = threadIdx.x >> 5;
    const int lane = threadIdx.x & 31;
    const int half = lane >> 4;
    const int mm   = lane & 15;

    const int mt = blockIdx.x * 4 + (wave >> 1);   // 4096 m-tiles total
    const int nt = wave & 1;                       // 2 n-tiles
    const int row0 = mt * 16;
    const int n0   = nt * 16;

    const float* Arow = inp  + (size_t)(row0 + mm) * I_DIM + 2 * half;
    const float* Brow = wall + (size_t)(n0   + mm) * I_DIM + 2 * half;

    v8f acc = {};
#pragma unroll 8
    for (int k0 = 0; k0 < I_DIM; k0 += 4) {
        v2f a = *(const v2f*)(Arow + k0);
        v2f b = *(const v2f*)(Brow + k0);
        acc = __builtin_amdgcn_wmma_f32_16x16x4_f32(
            /*neg_a=*/false, a, /*neg_b=*/false, b,
            /*c_mod=*/(short)0, acc, /*reuse_a=*/false, /*reuse_b=*/false);
    }

    // D layout: vgpr r -> row M = r + 8*half ; column N = lane & 15
    float* drow = angx + (size_t)(row0 + 8 * half) * NROWS + n0 + mm;
#pragma unroll
    for (int r = 0; r < 8; ++r)
        drow[(size_t)r * NROWS] = acc[r];
}

// ---------------------------------------------------------------------------
// 3) Serial scan: thread b owns sample b; states are scalars (rank-1).
//    expval = cos^2(angle_q0);  f,i,o = sigmoid(e), g = tanh(e)
// ---------------------------------------------------------------------------
__global__ __launch_bounds__(256) void scan_kernel(
    const float* __restrict__ angx,     // [65536][32]
    const float* __restrict__ sq,       // [32]
    const float* __restrict__ cq,       // [32]
    float* __restrict__ hs_hist,        // [T][B]
    float* __restrict__ cs_final)       // [B]
{
    const int b = threadIdx.x;
    float S[NG], C[NG];
#pragma unroll
    for (int g = 0; g < NG; ++g) { S[g] = sq[g * NQ]; C[g] = cq[g * NQ]; }

    float hs = 0.f, cs = 0.f;
    for (int t = 0; t < T_DIM; ++t) {
        const float* base = angx + ((size_t)t * B_DIM + b) * NROWS;
        if (t + 1 < T_DIM)
            __builtin_prefetch(base + (size_t)B_DIM * NROWS, 0, 0);

        const float af = base[0]  + hs * S[0] + C[0];
        const float ai = base[8]  + hs * S[1] + C[1];
        const float au = base[16] + hs * S[2] + C[2];
        const float ao = base[24] + hs * S[3] + C[3];

        float ef = cosf(af); ef *= ef;   // <Z0> = cos^2(theta0)
        float ei = cosf(ai); ei *= ei;
        float eu = cosf(au); eu *= eu;
        float eo = cosf(ao); eo *= eo;

        const float f  = 1.f / (1.f + expf(-ef));
        const float i_ = 1.f / (1.f + expf(-ei));
        const float g_ = tanhf(eu);
        const float o  = 1.f / (1.f + expf(-eo));

        cs = f * cs + i_ * g_;
        hs = o * tanhf(cs);
        hs_hist[(size_t)t * B_DIM + b] = hs;
    }
    cs_final[b] = cs;
}

// ---------------------------------------------------------------------------
// 4) Broadcast: outputs[t][b][:] = hs[t][b]; then hx[b][:], cx[b][:] tails.
//    float4 stores, exactly out_size = (T*B + 2*B) * H floats.
// ---------------------------------------------------------------------------
__global__ __launch_bounds__(256) void bcast_kernel(
    const float* __restrict__ hs_hist,
    const float* __restrict__ cs_final,
    float4* __restrict__ out)
{
    const size_t idx = (size_t)blockIdx.x * 256 + threadIdx.x;  // float4 index
    const size_t row = idx >> 6;                                // H/4 = 64 f4 per row
    float v;
    if (row < (size_t)MTOT) {
        v = hs_hist[row];
    } else if (row < (size_t)MTOT + B_DIM) {
        v = hs_hist[(size_t)(T_DIM - 1) * B_DIM + (row - MTOT)];  // hx
    } else {
        v = cs_final[row - MTOT - B_DIM];                          // cx
    }
    out[idx] = make_float4(v, v, v, v);
}

// ---------------------------------------------------------------------------
extern "C" void kernel_launch(void* const* d_in, const int* in_sizes, int n_in,
                              void* d_out, int out_size, void* d_ws, size_t ws_size,
                              hipStream_t stream) {
    (void)in_sizes; (void)n_in; (void)out_size; (void)ws_size;
    const float* inp = (const float*)d_in[0];
    float* ws = (float*)d_ws;

    prep_kernel<<<NG, 256, 0, stream>>>(
        (const float*)d_in[1],  (const float*)d_in[2],  (const float*)d_in[3],  (const float*)d_in[4],
        (const float*)d_in[5],  (const float*)d_in[6],  (const float*)d_in[7],  (const float*)d_in[8],
        (const float*)d_in[9],  (const float*)d_in[10], (const float*)d_in[11], (const float*)d_in[12],
        (const float*)d_in[13], (const float*)d_in[14], (const float*)d_in[15], (const float*)d_in[16],
        ws + WS_WALL, ws + WS_SQ, ws + WS_CQ);

    angx_gemm_kernel<<<MTOT / 64, 256, 0, stream>>>(   // 1024 blocks, 8 waves each
        inp, ws + WS_WALL, ws + WS_ANGX);

    scan_kernel<<<1, 256, 0, stream>>>(
        ws + WS_ANGX, ws + WS_SQ, ws + WS_CQ, ws + WS_HS, ws + WS_CS);

    const size_t total_f4 = ((size_t)MTOT + 2 * B_DIM) * (H_DIM / 4);  // 4,227,072
    bcast_kernel<<<(unsigned)(total_f4 / 256), 256, 0, stream>>>(
        ws + WS_HS, ws + WS_CS, (float4*)d_out);
}